// LDE_58961311040249
// MI455X (gfx1250) — compile-verified
//
#include <hip/hip_runtime.h>

typedef float v2f __attribute__((ext_vector_type(2)));
typedef float v4f __attribute__((ext_vector_type(4)));
typedef float v8f __attribute__((ext_vector_type(8)));

#define B_     16
#define T_     2048
#define NIN    128
#define NOUT   64
#define TC     32      // T-rows per chunk held in LDS
#define CHUNKS 4       // chunks processed per block (accumulators persist)
#define XPITCH 132     // x tile pitch: mult-of-4 (16B aligned rows), bank stride 4
#define TPITCH 36      // wT pitch: mult-of-4, idx*36 mod 64 all distinct

// ---------------------------------------------------------------------------
// Fused: per (b, group of 4 T-chunks):
//   GEMM1 (v_wmma_f32_16x16x4_f32): dot = Xc(32x128) @ M(128x64)
//   softmax over 64 codewords; weights kept TRANSPOSED wT[k][t] so that
//   GEMM1-D stores, GEMM2-A loads, and column sums are all contiguous.
//   GEMM2 (v_wmma_f32_16x16x4_f32): E1 += w^T(64x32) @ Xc(32x128) [regs persist]
// Flush partials with native f32 global atomics.
// ---------------------------------------------------------------------------
__global__ __launch_bounds__(256) void lde_accum(const float* __restrict__ x,
                                                 const float* __restrict__ s,
                                                 const float* __restrict__ m,
                                                 float* __restrict__ e1,    // B*64*128
                                                 float* __restrict__ wacc)  // B*64
{
    __shared__ __align__(16) float mS[NIN * NOUT];   // m[i][k] row-major
    __shared__ __align__(16) float xS[TC * XPITCH];  // x chunk rows
    __shared__ __align__(16) float wT[NOUT * TPITCH];// dot/weights, [k][t]
    __shared__ float xxS[TC];
    __shared__ float mmS[NOUT];
    __shared__ float sS[NOUT];

    const int tid  = threadIdx.x;
    const int wave = tid >> 5;
    const int lane = tid & 31;
    const int half = lane >> 4;   // 0: lanes 0-15, 1: lanes 16-31
    const int idx  = lane & 15;

    const int b     = blockIdx.x >> 4;   // 16 chunk-groups per b
    const int cg    = blockIdx.x & 15;
    const int tbase = cg * (TC * CHUNKS);

    // Cache m once per block (vectorized b128); precompute ||m_k||^2; load s.
    for (int q = tid; q < (NIN * NOUT) / 4; q += 256)
        *(v4f*)&mS[q * 4] = *(const v4f*)&m[q * 4];
    if (tid < NOUT) sS[tid] = s[tid];
    __syncthreads();
    if (tid < NOUT) {
        float acc = 0.f;
        for (int i = 0; i < NIN; ++i) { float v = mS[i * NOUT + tid]; acc += v * v; }
        mmS[tid] = acc;
    }

    v8f acc2[4];
    for (int mt = 0; mt < 4; ++mt) acc2[mt] = (v8f){};
    float wlocal = 0.f;

    for (int cc = 0; cc < CHUNKS; ++cc) {
        const int t0 = tbase + cc * TC;
        __syncthreads();   // previous chunk's readers of xS/wT are done
        // Stage x chunk, float4-vectorized (global_load_b128).
        for (int q = tid; q < (TC * NIN) / 4; q += 256) {
            int t = q >> 5, i4 = (q & 31) * 4;
            *(v4f*)&xS[t * XPITCH + i4] =
                *(const v4f*)&x[(size_t)(b * T_ + t0 + t) * NIN + i4];
        }
        __syncthreads();
        if (tid < TC) {
            float a = 0.f;
            for (int i = 0; i < NIN; ++i) { float v = xS[tid * XPITCH + i]; a += v * v; }
            xxS[tid] = a;
        }
        __syncthreads();

        // --- GEMM1: 8 tiles (2x4), one 16x16 tile per wave, K=128 ---
        {
            const int trow = (wave >> 2) * 16;   // t-tile
            const int kcol = (wave & 3) * 16;    // codeword-tile
            v8f c = (v8f){};
            #pragma unroll
            for (int k = 0; k < NIN; k += 4) {
                // A 16x4 f32: lane pair K{0,1}/{2,3} -> contiguous, ds_load_b64
                v2f a = *(const v2f*)&xS[(trow + idx) * XPITCH + k + 2 * half];
                v2f bv;
                bv.x = mS[(k + 2 * half)     * NOUT + kcol + idx];
                bv.y = mS[(k + 2 * half + 1) * NOUT + kcol + idx];
                c = __builtin_amdgcn_wmma_f32_16x16x4_f32(
                        false, a, false, bv, (short)0, c, false, false);
            }
            // D: lane holds M = v+8*half (t-dim), N = idx (k-dim).
            // Transposed store: 8 contiguous t's per lane -> 2x ds_store_b128.
            const int base = (kcol + idx) * TPITCH + trow + 8 * half;
            *(v4f*)&wT[base]     = (v4f){c[0], c[1], c[2], c[3]};
            *(v4f*)&wT[base + 4] = (v4f){c[4], c[5], c[6], c[7]};
        }
        __syncthreads();

        // --- softmax per T-row (column of wT); threads walk consecutive t ---
        if (tid < TC) {
            const float xx = xxS[tid];
            float best = -3.4e38f;
            for (int k = 0; k < NOUT; ++k) {
                float dot = wT[k * TPITCH + tid];
                float dd  = (xx - 2.f * dot + mmS[k]) * sS[k];
                wT[k * TPITCH + tid] = dd;
                best = fmaxf(best, dd);
            }
            float sum = 0.f;
            for (int k = 0; k < NOUT; ++k) {
                float ev = __expf(wT[k * TPITCH + tid] - best);
                wT[k * TPITCH + tid] = ev;
                sum += ev;
            }
            float inv = 1.f / sum;
            for (int k = 0; k < NOUT; ++k) wT[k * TPITCH + tid] *= inv;
        }
        __syncthreads();

        // per-codeword sums: contiguous row scan of wT
        if (tid < NOUT) {
            float cs = 0.f;
            for (int t = 0; t < TC; ++t) cs += wT[tid * TPITCH + t];
            wlocal += cs;
        }

        // --- GEMM2: E1 += w^T(64x32) @ Xc(32x128); wave owns col-block n0 ---
        {
            const int n0 = wave * 16;
            #pragma unroll
            for (int mt = 0; mt < 4; ++mt) {
                const int m0 = mt * 16;
                v8f c = acc2[mt];
                #pragma unroll
                for (int t = 0; t < TC; t += 4) {
                    // A[kk][t] = wT[kk*TPITCH + t] -> contiguous pair, b64
                    v2f a = *(const v2f*)&wT[(m0 + idx) * TPITCH + t + 2 * half];
                    v2f bv;
                    bv.x = xS[(t + 2 * half)     * XPITCH + n0 + idx];
                    bv.y = xS[(t + 2 * half + 1) * XPITCH + n0 + idx];
                    c = __builtin_amdgcn_wmma_f32_16x16x4_f32(
                            false, a, false, bv, (short)0, c, false, false);
                }
                acc2[mt] = c;
            }
        }
    }
    __syncthreads();

    // Flush partials with native f32 global atomics (agent scope, relaxed).
    if (tid < NOUT)
        __hip_atomic_fetch_add(&wacc[b * NOUT + tid], wlocal,
                               __ATOMIC_RELAXED, __HIP_MEMORY_SCOPE_AGENT);
    {
        const int n0 = wave * 16;
        #pragma unroll
        for (int mt = 0; mt < 4; ++mt) {
            const int m0 = mt * 16;
            #pragma unroll
            for (int v = 0; v < 8; ++v) {
                int kk = m0 + v + 8 * half;
                __hip_atomic_fetch_add(&e1[((size_t)b * NOUT + kk) * NIN + n0 + idx],
                                       acc2[mt][v],
                                       __ATOMIC_RELAXED, __HIP_MEMORY_SCOPE_AGENT);
            }
        }
    }
}

// e[b,k,i] = (E1[b,k,i] - W[b,k] * m[i,k]) / T
__global__ __launch_bounds__(256) void lde_final(const float* __restrict__ e1,
                                                 const float* __restrict__ wacc,
                                                 const float* __restrict__ m,
                                                 float* __restrict__ out)
{
    int o = blockIdx.x * 256 + threadIdx.x;          // < B*64*128
    int b = o >> 13;
    int k = (o >> 7) & 63;
    int i = o & 127;
    out[o] = (e1[o] - wacc[b * NOUT + k] * m[i * NOUT + k]) * (1.0f / (float)T_);
}

extern "C" void kernel_launch(void* const* d_in, const int* in_sizes, int n_in,
                              void* d_out, int out_size, void* d_ws, size_t ws_size,
                              hipStream_t stream)
{
    (void)in_sizes; (void)n_in; (void)out_size; (void)ws_size;
    const float* x = (const float*)d_in[0];   // (16, 2048, 128)
    const float* s = (const float*)d_in[1];   // (64,)
    const float* m = (const float*)d_in[2];   // (128, 64)

    float* e1   = (float*)d_ws;                       // B*64*128 floats
    float* wacc = e1 + (size_t)B_ * NOUT * NIN;       // B*64 floats

    hipMemsetAsync(d_ws, 0,
                   ((size_t)B_ * NOUT * NIN + (size_t)B_ * NOUT) * sizeof(float),
                   stream);

    // 16 b's x 16 chunk-groups = 256 blocks, 256 threads (8 wave32s)
    lde_accum<<<B_ * 16, 256, 0, stream>>>(x, s, m, e1, wacc);
    // 131072 outputs / 256
    lde_final<<<(B_ * NOUT * NIN) / 256, 256, 0, stream>>>(e1, wacc, m, (float*)d_out);
}